// DPLayer_56264071577841
// MI455X (gfx1250) — compile-verified
//
#include <hip/hip_runtime.h>
#include <hip/hip_bf16.h>
#include <stdint.h>

// Soft-DTW antidiagonal DP, one 256-thread workgroup (8 wave32) per batch.
// CDNA5 paths: 258KB LDS/workgroup (320KB WGP LDS), async global->LDS staging
// (global_load_async_to_lds_b128 + s_wait_asynccnt), split work-group barrier
// (s_barrier_signal/-wait with s_wait_dscnt) to overlap the next cost-element
// LDS load with barrier latency. DP runs in log2e-scaled domain so the softmin
// dependent chain is sub->exp2 / log2->sub->add with no interior multiplies.

#define SS 256
#define F_LOG2E 1.4426950408889634f
#define F_LN2 0.6931471805599453f
#define BIGS (1000000000.0f * F_LOG2E)   // BIG in scaled domain

__device__ __forceinline__ float fexp2(float x) {
#if defined(__HIP_DEVICE_COMPILE__) && __has_builtin(__builtin_amdgcn_exp2f)
    return __builtin_amdgcn_exp2f(x);       // v_exp_f32 (2^x)
#elif defined(__HIP_DEVICE_COMPILE__)
    return exp2f(x);
#else
    return x;  // host pass: never executed
#endif
}

__device__ __forceinline__ float flog2(float x) {
#if defined(__HIP_DEVICE_COMPILE__) && __has_builtin(__builtin_amdgcn_logf)
    return __builtin_amdgcn_logf(x);        // v_log_f32 (log2 x)
#elif defined(__HIP_DEVICE_COMPILE__)
    return log2f(x);
#else
    return x;  // host pass: never executed
#endif
}

__global__ __launch_bounds__(256, 1)
void softdtw_diag_kernel(const float* __restrict__ in, float* __restrict__ out) {
    __shared__ float tile[SS * SS];   // 256 KB cost tile
    __shared__ float bufA[SS + 4];    // rolling diagonal buffers (scaled domain)
    __shared__ float bufB[SS + 4];    // index 0 = BIG sentinel (row -1)

    const int i = threadIdx.x;        // row owned by this thread
    const int b = blockIdx.x;         // batch
    const float* src = in + (size_t)b * (SS * SS);

    // ---- Stage the 256KB tile into LDS with CDNA5 async global->LDS copies.
    // Each of 256 threads copies 64 x 16B chunks (stride 4096B): fully coalesced.
    {
        unsigned lds = (unsigned)(uintptr_t)(&tile[0]) + (unsigned)(i * 16);
        const char* g = (const char*)src + i * 16;
        #pragma unroll 4
        for (int c = 0; c < 64; ++c) {
            asm volatile("global_load_async_to_lds_b128 %0, %1, off"
                         :: "v"(lds), "v"(g) : "memory");
            lds += 4096u;
            g += 4096;
        }
        asm volatile("s_wait_asynccnt 0" ::: "memory");
    }
    __syncthreads();   // all waves' async copies complete + visible

    // ---- DP state, carried scaled by log2(e):  xs = x * LOG2E.
    // vs1 = own D at k-1 (scaled)        (pred D[i][j-1])
    // bs  = neighbor D at k-1 (scaled)   (pred D[i-1][j])   -- LDS read
    // nbs = neighbor D at k-2 (scaled)   (pred D[i-1][j-1]) -- last step's bs
    float vs1 = (i == 0) ? tile[0] * F_LOG2E : BIGS;  // diagonal k=0
    float nbs = BIGS;                                 // diagonal k=-1
    bufA[i + 1] = vs1;
    if (i == 0) { bufA[0] = BIGS; bufB[0] = BIGS; }   // row -1 sentinels
    __syncthreads();

    float* rd = bufA;
    float* wr = bufB;

    // tile index of C[i][k-i] is i*255 + k; prefetch for k=1 (scaled).
    int idx = i * (SS - 1) + 1;
    float cs = tile[idx] * F_LOG2E;

    for (int k = 1; k <= 2 * SS - 2; ++k) {
        float bs = rd[i];                       // neighbor's D at k-1 (scaled)

        // softmin_gamma=1 in base-2 domain: no muls in the dependent chain.
        float ms = fminf(vs1, fminf(bs, nbs));
        float s  = fexp2(ms - vs1) + fexp2(ms - bs) + fexp2(ms - nbs);
        float vals = cs + ms - flog2(s);

        // mask cells outside the matrix (exactly BIG in natural units)
        bool valid = ((unsigned)(k - i)) <= (unsigned)(SS - 1);
        vals = valid ? vals : BIGS;

        wr[i + 1] = vals;

        // Split barrier: commit our LDS store, signal, then overlap the next
        // cost-element load with the barrier latency, then wait.
        asm volatile("s_wait_dscnt 0x0\n\t"
                     "s_barrier_signal -1" ::: "memory");
        ++idx;
        cs = tile[idx] * F_LOG2E;   // prefetch C[i][(k+1)-i]; last one unused,
                                    // lands in bufA[0] sentinel (in-allocation)
        asm volatile("s_barrier_wait -1" ::: "memory");

        float* t = rd; rd = wr; wr = t;
        nbs = bs;
        vs1 = vals;
    }

    // after k = 2S-2, thread 255 holds D[S-1][S-1] (scaled); rescale once.
    if (i == SS - 1) out[b] = vs1 * F_LN2;
}

extern "C" void kernel_launch(void* const* d_in, const int* in_sizes, int n_in,
                              void* d_out, int out_size, void* d_ws, size_t ws_size,
                              hipStream_t stream) {
    (void)n_in; (void)d_ws; (void)ws_size; (void)out_size;
    const float* in = (const float*)d_in[0];
    float* out = (float*)d_out;
    const int B = in_sizes[0] / (SS * SS);   // 512 batches
    softdtw_diag_kernel<<<dim3(B), dim3(256), 0, stream>>>(in, out);
}